// PhysicsInformedQuantumCircuit_9294309229151
// MI455X (gfx1250) — compile-verified
//
#include <hip/hip_runtime.h>

// ---------------------------------------------------------------------------
// 20-qubit state-vector simulator for MI455X (gfx1250).
//
// State: 2^20 complex64 amplitudes as split re/im fp32 planes living in d_out
// (final output layout is stack([re, im]) -> zero-copy finish).
//
// Per layer (all passes L2-resident: 8MB state << 192MB L2):
//   * groups 0..3: dense 16x16 complex unitary (Kron of four 2x2 gates)
//     applied with V_WMMA_F32_16X16X4_F32 (fp32 matrix engine == complex64
//     numerics). Complex product = 4 real WMMAs, K-loop of 4 -> 16 WMMA/tile.
//   * group 4 pass additionally applies all 20 fused RZZ diagonal phases
//     (diagonal commutes with nothing pending -> fused into final store).
// Normalization scale is folded into the layer-0/group-0 unitary; the first
// apply pass reads d_in directly, so there is no separate normalize pass.
// Reduction is a fixed-order two-stage tree (deterministic, no atomics).
// ---------------------------------------------------------------------------

#define NQ      20
#define DEPTH   8
#define DIM     (1 << NQ)
#define PSTRIDE 80           // 3*20 rotation params + 20 zz params per layer
#define NGRP    5            // 5 groups of 4 qubits
#define RBLKS   1024         // partial-sum blocks in reduction stage 1

typedef float v2f __attribute__((ext_vector_type(2)));
typedef float v8f __attribute__((ext_vector_type(8)));

// ---------------- deterministic norm reduction ----------------

__global__ void qc_reduce1(const float* __restrict__ re,
                           const float* __restrict__ im,
                           float* __restrict__ part) {
  __shared__ float sm[256];
  float s = 0.0f;
  for (int i = blockIdx.x * blockDim.x + threadIdx.x; i < DIM;
       i += gridDim.x * blockDim.x) {
    const float a = re[i], b = im[i];
    s += a * a + b * b;
  }
  sm[threadIdx.x] = s;
  __syncthreads();
  for (int o = 128; o > 0; o >>= 1) {
    if (threadIdx.x < o) sm[threadIdx.x] += sm[threadIdx.x + o];
    __syncthreads();
  }
  if (threadIdx.x == 0) part[blockIdx.x] = sm[0];
}

__global__ void qc_reduce2(const float* __restrict__ part,
                           float* __restrict__ acc) {
  __shared__ float sm[256];
  const int t = threadIdx.x;
  // fixed-order accumulation: deterministic across replays
  float s = part[t] + part[t + 256] + part[t + 512] + part[t + 768];
  sm[t] = s;
  __syncthreads();
  for (int o = 128; o > 0; o >>= 1) {
    if (t < o) sm[t] += sm[t + o];
    __syncthreads();
  }
  if (t == 0) acc[0] = sm[0];
}

// ---------------- build 16x16 group unitaries (Kron of four 2x2) ------------
// U = Rz(tz) @ Ry(ty) @ Rx(tx) per qubit; group gate = u_q0 (x) ... (x) u_q3
// with q0 = most-significant bit of the 4-bit group index.
// Layer 0 / group 0 unitary absorbs the 1/sqrt(norm) normalization scale.

__global__ void qc_build_gates(const float* __restrict__ gp,
                               const float* __restrict__ acc,
                               float* __restrict__ Ure,
                               float* __restrict__ Uim) {
  const int layer = blockIdx.x / NGRP;
  const int g     = blockIdx.x % NGRP;
  const int t = threadIdx.x;          // 0..255
  const int r = t >> 4, s = t & 15;
  const float* p = gp + layer * PSTRIDE;

  float ur[4][2][2], ui[4][2][2];
#pragma unroll
  for (int k = 0; k < 4; ++k) {
    const int q = 4 * g + k;
    const float tx = p[3 * q], ty = p[3 * q + 1], tz = p[3 * q + 2];
    const float cx = cosf(0.5f * tx), sx = sinf(0.5f * tx);
    const float cy = cosf(0.5f * ty), sy = sinf(0.5f * ty);
    const float cz = cosf(0.5f * tz), sz = sinf(0.5f * tz);
    // M = Ry*Rx
    const float m00r =  cy * cx, m00i =  sy * sx;
    const float m01r = -sy * cx, m01i = -cy * sx;
    const float m10r =  sy * cx, m10i = -cy * sx;
    const float m11r =  cy * cx, m11i = -sy * sx;
    // row0 *= (cz - i sz); row1 *= (cz + i sz)
    ur[k][0][0] = cz * m00r + sz * m00i;  ui[k][0][0] = cz * m00i - sz * m00r;
    ur[k][0][1] = cz * m01r + sz * m01i;  ui[k][0][1] = cz * m01i - sz * m01r;
    ur[k][1][0] = cz * m10r - sz * m10i;  ui[k][1][0] = cz * m10i + sz * m10r;
    ur[k][1][1] = cz * m11r - sz * m11i;  ui[k][1][1] = cz * m11i + sz * m11r;
  }
  float vr = 1.0f, vi = 0.0f;
#pragma unroll
  for (int k = 0; k < 4; ++k) {
    const int br = (r >> (3 - k)) & 1, bs = (s >> (3 - k)) & 1;
    const float wr = ur[k][br][bs], wi = ui[k][br][bs];
    const float nr = vr * wr - vi * wi;
    const float ni = vr * wi + vi * wr;
    vr = nr; vi = ni;
  }
  const float scale = (layer == 0 && g == 0) ? (1.0f / sqrtf(acc[0])) : 1.0f;
  const int off = ((layer * NGRP + g) << 8) + t;
  Ure[off] = vr * scale;
  Uim[off] = vi * scale;
}

// ---------------- WMMA group-gate application ----------------
// A-matrix VGPR layout (ISA 7.12.2, 32-bit A 16x4): lane<16 -> M=lane, K=v;
// lane>=16 -> M=lane-16, K=v+2. B/C/D: row striped across lanes, rows v & v+half.

// Groups 0..3: group bits at positions p..p+3 (p = 16-4g >= 4).
// Tile: Y = U * X, X rows at stride 2^p, 16 contiguous columns.
// src/dst may alias (in-place: tiles are independent) or differ (first pass).
__global__ void qc_apply_group_hi(const float* __restrict__ src_re,
                                  const float* __restrict__ src_im,
                                  float* __restrict__ dst_re,
                                  float* __restrict__ dst_im,
                                  const float* __restrict__ Ure,
                                  const float* __restrict__ Uim, int p) {
  const int lane = threadIdx.x & 31;
  const int half = lane >> 4;
  const int lr   = lane & 15;
  const unsigned tile   = blockIdx.x * (blockDim.x >> 5) + (threadIdx.x >> 5);
  const unsigned stride = 1u << p;
  const unsigned lmask  = (1u << (p - 4)) - 1u;
  const unsigned hi     = tile >> (p - 4);
  const unsigned lo     = (tile & lmask) << 4;
  const unsigned base   = (hi << (p + 4)) + lo;

  v2f aur[4], aui[4], auin[4], bre[4], bim[4];
#pragma unroll
  for (int c = 0; c < 4; ++c) {
    const int k0 = 4 * c + 2 * half;
    aur[c][0] = Ure[lr * 16 + k0];      aur[c][1] = Ure[lr * 16 + k0 + 1];
    aui[c][0] = Uim[lr * 16 + k0];      aui[c][1] = Uim[lr * 16 + k0 + 1];
    auin[c] = -aui[c];
    bre[c][0] = src_re[base + (unsigned)k0 * stride + lr];
    bre[c][1] = src_re[base + (unsigned)(k0 + 1) * stride + lr];
    bim[c][0] = src_im[base + (unsigned)k0 * stride + lr];
    bim[c][1] = src_im[base + (unsigned)(k0 + 1) * stride + lr];
  }
  v8f accr = {}, acci = {};
#pragma unroll
  for (int c = 0; c < 4; ++c) {
    // Y_re = U_re*X_re + (-U_im)*X_im ; Y_im = U_re*X_im + U_im*X_re
    accr = __builtin_amdgcn_wmma_f32_16x16x4_f32(false, aur[c],  false, bre[c], (short)0, accr, false, false);
    accr = __builtin_amdgcn_wmma_f32_16x16x4_f32(false, auin[c], false, bim[c], (short)0, accr, false, false);
    acci = __builtin_amdgcn_wmma_f32_16x16x4_f32(false, aur[c],  false, bim[c], (short)0, acci, false, false);
    acci = __builtin_amdgcn_wmma_f32_16x16x4_f32(false, aui[c],  false, bre[c], (short)0, acci, false, false);
  }
#pragma unroll
  for (int v = 0; v < 8; ++v) {
    const unsigned row  = (unsigned)v + 8u * (unsigned)half;
    const unsigned addr = base + row * stride + lr;
    dst_re[addr] = accr[v];
    dst_im[addr] = acci[v];
  }
}

// Group 4: group bits are the lowest 4 bits (p=0). Compute Y^T = X^T * U^T
// (state tiles = A, rows contiguous; U = B). This is the last 1q pass of the
// layer, so the fused RZZ diagonal phase is applied at the store.
// RZZ pairs: (q,q+1) q=0..18 -> adjacent index bits (19-k,18-k); plus (0,19)
// -> bits (19,0). angle(i) = -0.5 * sum_k theta_k * (1 - 2*xor(bits)).
__global__ void qc_apply_group_low_zz(float* __restrict__ sre,
                                      float* __restrict__ sim,
                                      const float* __restrict__ Ure,
                                      const float* __restrict__ Uim,
                                      const float* __restrict__ th) {
  const int lane = threadIdx.x & 31;
  const int half = lane >> 4;
  const int lr   = lane & 15;
  const unsigned tile = blockIdx.x * (blockDim.x >> 5) + (threadIdx.x >> 5);
  const unsigned base = tile << 8;

  float t[20];
#pragma unroll
  for (int k = 0; k < 20; ++k) t[k] = th[k];   // uniform scalar loads

  v2f asr[4], asi[4], bur[4], bui[4], buin[4];
#pragma unroll
  for (int c = 0; c < 4; ++c) {
    const int k0 = 4 * c + 2 * half;
    asr[c][0] = sre[base + lr * 16 + k0];  asr[c][1] = sre[base + lr * 16 + k0 + 1];
    asi[c][0] = sim[base + lr * 16 + k0];  asi[c][1] = sim[base + lr * 16 + k0 + 1];
    // B = U^T : B[K][N] = U[N][K] -> same row-major indexing expression
    bur[c][0] = Ure[lr * 16 + k0];         bur[c][1] = Ure[lr * 16 + k0 + 1];
    bui[c][0] = Uim[lr * 16 + k0];         bui[c][1] = Uim[lr * 16 + k0 + 1];
    buin[c] = -bui[c];
  }
  v8f accr = {}, acci = {};
#pragma unroll
  for (int c = 0; c < 4; ++c) {
    // Y^T_re = X^T_re*U^T_re + X^T_im*(-U^T_im) ; Y^T_im = X^T_re*U^T_im + X^T_im*U^T_re
    accr = __builtin_amdgcn_wmma_f32_16x16x4_f32(false, asr[c], false, bur[c],  (short)0, accr, false, false);
    accr = __builtin_amdgcn_wmma_f32_16x16x4_f32(false, asi[c], false, buin[c], (short)0, accr, false, false);
    acci = __builtin_amdgcn_wmma_f32_16x16x4_f32(false, asr[c], false, bui[c],  (short)0, acci, false, false);
    acci = __builtin_amdgcn_wmma_f32_16x16x4_f32(false, asi[c], false, bur[c],  (short)0, acci, false, false);
  }

  // Per-lane output indices differ only in bits 4..7 (m = v + 8*half).
  // Pairs k=0..10 and k=16..19 touch only bits {0..3, 8..19}: fixed per lane.
  // Pairs k=11..15 touch bits 4..8: recomputed per output element.
  const unsigned i0 = base + (unsigned)lr;     // bits 4..7 zero; unused by fixed pairs
  float sfix = 0.0f;
#pragma unroll
  for (int k = 0; k <= 10; ++k) {
    const unsigned xb = ((i0 >> (19 - k)) ^ (i0 >> (18 - k))) & 1u;
    sfix += xb ? -t[k] : t[k];
  }
#pragma unroll
  for (int k = 16; k <= 18; ++k) {
    const unsigned xb = ((i0 >> (19 - k)) ^ (i0 >> (18 - k))) & 1u;
    sfix += xb ? -t[k] : t[k];
  }
  {
    const unsigned xb = ((i0 >> 19) ^ i0) & 1u;   // wraparound pair (0,19)
    sfix += xb ? -t[19] : t[19];
  }

#pragma unroll
  for (int v = 0; v < 8; ++v) {
    const unsigned m    = (unsigned)v + 8u * (unsigned)half;
    const unsigned addr = base + m * 16u + lr;
    float sv = sfix;
#pragma unroll
    for (int k = 11; k <= 15; ++k) {
      const unsigned xb = ((addr >> (19 - k)) ^ (addr >> (18 - k))) & 1u;
      sv += xb ? -t[k] : t[k];
    }
    const float ang = -0.5f * sv;
    float sn, cs;
    __sincosf(ang, &sn, &cs);
    const float yr = accr[v], yi = acci[v];
    sre[addr] = yr * cs - yi * sn;
    sim[addr] = yr * sn + yi * cs;
  }
}

// ---------------- launch ----------------

extern "C" void kernel_launch(void* const* d_in, const int* in_sizes, int n_in,
                              void* d_out, int out_size, void* d_ws, size_t ws_size,
                              hipStream_t stream) {
  const float* in_re = (const float*)d_in[0];
  const float* in_im = (const float*)d_in[1];
  const float* gp    = (const float*)d_in[2];

  float* out = (float*)d_out;
  float* sre = out;              // output re plane doubles as working state
  float* sim = out + DIM;        // output im plane

  float* ws   = (float*)d_ws;
  float* part = ws;                              // [RBLKS] partial sums
  float* acc  = ws + RBLKS;                      // [1] norm^2
  float* Ure  = ws + RBLKS + 16;                 // [DEPTH*5*256] unitary re
  float* Uim  = Ure + DEPTH * NGRP * 256;        // [DEPTH*5*256] unitary im

  qc_reduce1<<<RBLKS, 256, 0, stream>>>(in_re, in_im, part);
  qc_reduce2<<<1, 256, 0, stream>>>(part, acc);
  qc_build_gates<<<DEPTH * NGRP, 256, 0, stream>>>(gp, acc, Ure, Uim);

  for (int layer = 0; layer < DEPTH; ++layer) {
    for (int g = 0; g < NGRP; ++g) {
      const float* ur = Ure + ((layer * NGRP + g) << 8);
      const float* ui = Uim + ((layer * NGRP + g) << 8);
      if (g < 4) {
        // first pass reads the (unnormalized) input; scale lives in U(0,0)
        const float* sr = (layer == 0 && g == 0) ? in_re : sre;
        const float* si = (layer == 0 && g == 0) ? in_im : sim;
        qc_apply_group_hi<<<512, 256, 0, stream>>>(sr, si, sre, sim, ur, ui,
                                                   16 - 4 * g);
      } else {
        qc_apply_group_low_zz<<<512, 256, 0, stream>>>(
            sre, sim, ur, ui, gp + layer * PSTRIDE + 60);
      }
    }
  }
}